// EarthMoversDistanceLoss_44195213476247
// MI455X (gfx1250) — compile-verified
//
#include <hip/hip_runtime.h>
#include <hip/hip_bf16.h>
#include <stdint.h>

// EMD loss: out[b] = sum_j (cumsum(x)[b,j] - cumsum(y)[b,j])^2
// = streaming prefix-scan of (x - y) per row + sum of squares.
// Memory-bound: 256 MB traffic -> ~11.5 us at 23.3 TB/s. No WMMA (matmul
// formulation would be ~548 GFLOP for the same bytes). CDNA5 feature used:
// async global->LDS loads (ASYNCcnt) for double-buffered streaming.

#define BINS    4096
#define NWAVES  8
#define SEG     (BINS / NWAVES)   // 512 elements per wave
#define CHUNK   128               // 32 lanes * float4 per iteration
#define NITERS  (SEG / CHUNK)     // 4

#if defined(__HIP_DEVICE_COMPILE__) && __has_builtin(__builtin_amdgcn_global_load_async_to_lds_b128)
#define EMD_ASYNC 1
#else
#define EMD_ASYNC 0
#endif

#if EMD_ASYNC
// Builtin signature (from clang diagnostic):
//   void __builtin_amdgcn_global_load_async_to_lds_b128(
//       int4 addrspace(1)* src, int4 addrspace(3)* dst, imm int off, imm int cpol)
typedef __attribute__((ext_vector_type(4))) int i32x4;
typedef __attribute__((address_space(1))) i32x4 gbl_i32x4;
typedef __attribute__((address_space(3))) i32x4 lds_i32x4;

static __device__ __forceinline__ gbl_i32x4* emd_to_gbl(const void* p) {
    return (gbl_i32x4*)(uint64_t)(uintptr_t)p;
}
// Generic LDS pointer: low 32 bits are the LDS byte offset (aperture base has
// zero low bits), so truncation yields a valid AS(3) pointer.
static __device__ __forceinline__ lds_i32x4* emd_to_lds(const void* p) {
    return (lds_i32x4*)(uint32_t)(uint64_t)(uintptr_t)p;
}

template <int N>
static __device__ __forceinline__ void emd_wait_async() {
#if __has_builtin(__builtin_amdgcn_s_wait_asynccnt)
    __builtin_amdgcn_s_wait_asynccnt(N);
#else
    asm volatile("s_wait_asynccnt %0" :: "i"(N) : "memory");
#endif
}
#endif  // EMD_ASYNC

__global__ __launch_bounds__(256) void emd_kernel(const float* __restrict__ x,
                                                  const float* __restrict__ y,
                                                  float* __restrict__ out) {
    const int row  = blockIdx.x;
    const int lane = threadIdx.x & 31;
    const int w    = threadIdx.x >> 5;

    const float* xr = x + (size_t)row * BINS + w * SEG + lane * 4;
    const float* yr = y + (size_t)row * BINS + w * SEG + lane * 4;

    __shared__ float sA[NWAVES];  // sum of squared local-prefix values
    __shared__ float sB[NWAVES];  // sum of local-prefix values
    __shared__ float sT[NWAVES];  // segment total

    float accA = 0.f, accB = 0.f, carry = 0.f;

#if EMD_ASYNC
    // [wave][buf][x/y][lane] staging: 8*2*2*32*16B = 16 KB
    __shared__ float4 stage[NWAVES][2][2][32];
    // Prologue: kick off iteration-0 loads (512 B per instr per wave).
    __builtin_amdgcn_global_load_async_to_lds_b128(
        emd_to_gbl(xr), emd_to_lds(&stage[w][0][0][lane]), 0, 0);
    __builtin_amdgcn_global_load_async_to_lds_b128(
        emd_to_gbl(yr), emd_to_lds(&stage[w][0][1][lane]), 0, 0);
#endif

#pragma unroll
    for (int it = 0; it < NITERS; ++it) {
        float4 xv, yv;
#if EMD_ASYNC
        const int buf = it & 1;
        if (it + 1 < NITERS) {
            __builtin_amdgcn_global_load_async_to_lds_b128(
                emd_to_gbl(xr + (it + 1) * CHUNK),
                emd_to_lds(&stage[w][buf ^ 1][0][lane]), 0, 0);
            __builtin_amdgcn_global_load_async_to_lds_b128(
                emd_to_gbl(yr + (it + 1) * CHUNK),
                emd_to_lds(&stage[w][buf ^ 1][1][lane]), 0, 0);
            emd_wait_async<2>();  // oldest 2 (current buf) complete, next 2 in flight
        } else {
            emd_wait_async<0>();
        }
        xv = stage[w][buf][0][lane];
        yv = stage[w][buf][1][lane];
#else
        xv = *(const float4*)(xr + it * CHUNK);
        yv = *(const float4*)(yr + it * CHUNK);
#endif
        // In-lane inclusive prefix of z = x - y over 4 elements.
        float p0 = xv.x - yv.x;
        float p1 = p0 + (xv.y - yv.y);
        float p2 = p1 + (xv.z - yv.z);
        float p3 = p2 + (xv.w - yv.w);

        // Wave-wide inclusive scan of lane totals (wave32: 5 shuffle steps).
        float s = p3;
#pragma unroll
        for (int d = 1; d < 32; d <<= 1) {
            float n = __shfl_up(s, d, 32);
            if (lane >= d) s += n;
        }
        float base = carry + (s - p3);  // exclusive prefix for this lane
        float v0 = p0 + base, v1 = p1 + base, v2 = p2 + base, v3 = p3 + base;
        accA += v0 * v0 + v1 * v1 + v2 * v2 + v3 * v3;
        accB += v0 + v1 + v2 + v3;
        carry += __shfl(s, 31, 32);     // chunk total -> running carry
    }

    // Wave-reduce A and B (butterfly).
#pragma unroll
    for (int m = 16; m >= 1; m >>= 1) {
        accA += __shfl_xor(accA, m, 32);
        accB += __shfl_xor(accB, m, 32);
    }
    if (lane == 0) { sA[w] = accA; sB[w] = accB; sT[w] = carry; }
    __syncthreads();

    // Cross-wave combine in wave 0:
    // row = sum_w [ A_w + 2*c_w*B_w + SEG*c_w^2 ], c_w = excl-prefix of T.
    if (threadIdx.x < 32) {
        float Aw = 0.f, Bw = 0.f, Tw = 0.f;
        if (lane < NWAVES) { Aw = sA[lane]; Bw = sB[lane]; Tw = sT[lane]; }
        float sc = Tw;
#pragma unroll
        for (int d = 1; d < NWAVES; d <<= 1) {
            float n = __shfl_up(sc, d, 32);
            if (lane >= d) sc += n;
        }
        float c = sc - Tw;
        float term = (lane < NWAVES) ? (Aw + 2.f * c * Bw + (float)SEG * c * c)
                                     : 0.f;
#pragma unroll
        for (int m = NWAVES / 2; m >= 1; m >>= 1)
            term += __shfl_xor(term, m, 32);
        if (lane == 0) out[row] = term;
    }
}

extern "C" void kernel_launch(void* const* d_in, const int* in_sizes, int n_in,
                              void* d_out, int out_size, void* d_ws, size_t ws_size,
                              hipStream_t stream) {
    const float* x = (const float*)d_in[0];
    const float* y = (const float*)d_in[1];
    float* out = (float*)d_out;
    const int B = out_size;  // 8192 rows, one block per row
    emd_kernel<<<B, 256, 0, stream>>>(x, y, out);
    (void)in_sizes; (void)n_in; (void)d_ws; (void)ws_size;
}